// GNN_54322746360585
// MI455X (gfx1250) — compile-verified
//
#include <hip/hip_runtime.h>
#include <math.h>

#define N_NODES   50000
#define N_EDGES   800000
#define TOT_EDGES (N_EDGES + N_NODES)   // self-loops appended
#define N_GRAPHS  64
#define HEADS     8
#define NEG_SLOPE 0.2f
#define BN_EPS    1e-5f

typedef float v2f __attribute__((ext_vector_type(2)));
typedef float v8f __attribute__((ext_vector_type(8)));

static inline int cdiv_i(int a, int b) { return (a + b - 1) / b; }

// ---- monotone float<->uint key for atomicMax on floats ----
__device__ __forceinline__ unsigned f2key(float f) {
    unsigned v = __float_as_uint(f);
    return (v & 0x80000000u) ? ~v : (v | 0x80000000u);
}
__device__ __forceinline__ float key2f(unsigned k) {
    unsigned v = (k & 0x80000000u) ? (k & 0x7FFFFFFFu) : ~k;
    return __uint_as_float(v);
}

__device__ __forceinline__ void edge_sd(const int* __restrict__ ei, int e, int& s, int& d) {
    if (e < N_EDGES) { s = ei[e]; d = ei[N_EDGES + e]; }
    else             { s = e - N_EDGES; d = s; }
}

// ------------------------------------------------------------------
__global__ void gnn_zero(float* __restrict__ p, int n) {
    int i = blockIdx.x * blockDim.x + threadIdx.x;
    if (i < n) p[i] = 0.0f;
}

// ---- FP32 WMMA GEMM: H[N,F] = A[N,K] @ W[K,F] ----
// One wave computes a 16x64 output strip: 4 accumulators share one A fragment
// per k-step (4 x v_wmma_f32_16x16x4_f32 per A load). Pointer-increment
// addressing keeps the inner loop free of 64-bit index multiplies. Row index
// is clamped (not branch-guarded) so the A load is an unconditional b64 and
// the loop carries no EXEC manipulation; stores remain masked.
__global__ void gemm_wmma_f32(const float* __restrict__ A, const float* __restrict__ W,
                              float* __restrict__ H, int N, int K, int F) {
    int lane   = threadIdx.x & 31;
    int wave   = threadIdx.x >> 5;
    int strips = F >> 6;                         // 64-column strips
    int tilesM = (N + 15) >> 4;
    int strip  = blockIdx.x * (blockDim.x >> 5) + wave;
    if (strip >= tilesM * strips) return;        // wave-uniform exit
    int tm = strip / strips;
    int tn = (strip - tm * strips) << 6;         // first column of strip

    int halfK = (lane >> 4) << 1;                // lanes 0-15 -> K0,K1 ; 16-31 -> K2,K3
    int row   = tm * 16 + (lane & 15);
    if (row >= N) row = N - 1;                   // clamp: out-of-range rows masked at store
    int col   = tn + (lane & 15);

    const float* ap = A + (size_t)row * K + halfK;
    const float* wp = W + (size_t)halfK * F + col;
    const size_t wstep = (size_t)4 * F;

    v8f acc0 = {}, acc1 = {}, acc2 = {}, acc3 = {};
    for (int k = 0; k < K; k += 4) {
        v2f a = *(const v2f*)ap;                 // unconditional global_load_b64
        ap += 4;

        __builtin_prefetch(wp + 2 * wstep, 0, 1);   // global_prefetch_b8 (W panel ahead)

        v2f b0, b1, b2, b3;
        b0.x = wp[0];  b0.y = wp[(size_t)F + 0];
        b1.x = wp[16]; b1.y = wp[(size_t)F + 16];
        b2.x = wp[32]; b2.y = wp[(size_t)F + 32];
        b3.x = wp[48]; b3.y = wp[(size_t)F + 48];
        wp += wstep;

        acc0 = __builtin_amdgcn_wmma_f32_16x16x4_f32(false, a, false, b0, (short)0, acc0, false, false);
        acc1 = __builtin_amdgcn_wmma_f32_16x16x4_f32(false, a, false, b1, (short)0, acc1, false, false);
        acc2 = __builtin_amdgcn_wmma_f32_16x16x4_f32(false, a, false, b2, (short)0, acc2, false, false);
        acc3 = __builtin_amdgcn_wmma_f32_16x16x4_f32(false, a, false, b3, (short)0, acc3, false, false);
    }

    // C/D layout: VGPR j -> M = j (+8 for upper half-wave), N = lane&15
    int baseM = tm * 16 + ((lane >> 4) << 3);
#pragma unroll
    for (int j = 0; j < 8; ++j) {
        int m = baseM + j;
        if (m < N) {
            float* hp = H + (size_t)m * F + col;
            hp[0]  = acc0[j];
            hp[16] = acc1[j];
            hp[32] = acc2[j];
            hp[48] = acc3[j];
        }
    }
}

// ---- per-(node,head) attention logits: als = <h, a_src>, ald = <h, a_dst> ----
__global__ void attn_logits(const float* __restrict__ H, const float* __restrict__ as_,
                            const float* __restrict__ ad_, float* __restrict__ als,
                            float* __restrict__ ald, int C) {
    int i = blockIdx.x * blockDim.x + threadIdx.x;
    if (i >= N_NODES * HEADS) return;
    int n = i / HEADS, h = i - n * HEADS;
    int F = HEADS * C;
    const float* hp = H + (size_t)n * F + h * C;
    const float* sp = as_ + h * C;
    const float* dp = ad_ + h * C;
    float ss = 0.0f, dd = 0.0f;
    for (int c = 0; c < C; ++c) { float v = hp[c]; ss += v * sp[c]; dd += v * dp[c]; }
    als[i] = ss;
    ald[i] = dd;
}

// ---- pass 1 over edges: segment max of leaky-relu logits into dst ----
__global__ void edge_max(const int* __restrict__ ei, const float* __restrict__ als,
                         const float* __restrict__ ald, unsigned* __restrict__ mkey) {
    int i = blockIdx.x * blockDim.x + threadIdx.x;
    if (i >= TOT_EDGES * HEADS) return;
    int e = i >> 3, h = i & 7;
    int s, d; edge_sd(ei, e, s, d);
    float v = als[s * HEADS + h] + ald[d * HEADS + h];
    v = (v > 0.0f) ? v : NEG_SLOPE * v;
    atomicMax(&mkey[d * HEADS + h], f2key(v));
}

// ---- pass 2 over edges: acc[dst] += exp(e - m[dst]) * h[src]; s[dst] += exp ----
__global__ void edge_accum(const int* __restrict__ ei, const float* __restrict__ als,
                           const float* __restrict__ ald, const unsigned* __restrict__ mkey,
                           float* __restrict__ ssum, const float* __restrict__ H,
                           float* __restrict__ acc, int C) {
    int lane = threadIdx.x & 31;
    int e = blockIdx.x * (blockDim.x >> 5) + (threadIdx.x >> 5);
    if (e >= TOT_EDGES) return;                    // wave-uniform
    int s, d; edge_sd(ei, e, s, d);
    int F = HEADS * C;
    const float* hs = H + (size_t)s * F;
    float*       ad = acc + (size_t)d * F;
#pragma unroll
    for (int h = 0; h < HEADS; ++h) {
        float v = als[s * HEADS + h] + ald[d * HEADS + h];
        v = (v > 0.0f) ? v : NEG_SLOPE * v;
        float ee = expf(v - key2f(mkey[d * HEADS + h]));
        if (lane == 0) atomicAdd(&ssum[d * HEADS + h], ee);
        for (int c = lane; c < C; c += 32)
            atomicAdd(&ad[h * C + c], ee * hs[h * C + c]);
    }
}

// ---- per-(node,feature): softmax-normalize, +bias, BN(eval), ReLU ----
__global__ void gat_finalize(const float* __restrict__ acc, const float* __restrict__ ssum,
                             const float* __restrict__ bias, const float* __restrict__ gamma,
                             const float* __restrict__ beta, const float* __restrict__ mean,
                             const float* __restrict__ var, float* __restrict__ out, int C) {
    int F = HEADS * C;
    int i = blockIdx.x * blockDim.x + threadIdx.x;
    if (i >= N_NODES * F) return;
    int n = i / F, f = i - n * F;
    int h = f / C;
    float v = acc[i] / (ssum[n * HEADS + h] + 1e-16f) + bias[f];
    v = (v - mean[f]) * rsqrtf(var[f] + BN_EPS) * gamma[f] + beta[f];
    out[i] = fmaxf(v, 0.0f);
}

// ---- gate = h3 @ gw + gb (wave/node), plus per-graph max of gate ----
__global__ void gate_kernel(const float* __restrict__ h3, const float* __restrict__ gw,
                            const float* __restrict__ gb, const int* __restrict__ batch,
                            float* __restrict__ gate, unsigned* __restrict__ gmkey) {
    int lane = threadIdx.x & 31;
    int n = blockIdx.x * (blockDim.x >> 5) + (threadIdx.x >> 5);
    if (n >= N_NODES) return;
    float v = 0.0f;
    for (int f = lane; f < 256; f += 32) v += h3[(size_t)n * 256 + f] * gw[f];
    for (int off = 16; off > 0; off >>= 1) v += __shfl_xor(v, off, 32);
    if (lane == 0) {
        float g = v + gb[0];
        gate[n] = g;
        atomicMax(&gmkey[batch[n]], f2key(g));
    }
}

// ---- pooled[g] += exp(gate - gm) * h3[n]; gs[g] += exp ----
__global__ void pool_kernel(const float* __restrict__ h3, const float* __restrict__ gate,
                            const unsigned* __restrict__ gmkey, const int* __restrict__ batch,
                            float* __restrict__ gs, float* __restrict__ pooled) {
    int lane = threadIdx.x & 31;
    int n = blockIdx.x * (blockDim.x >> 5) + (threadIdx.x >> 5);
    if (n >= N_NODES) return;
    int b = batch[n];
    float ge = expf(gate[n] - key2f(gmkey[b]));
    if (lane == 0) atomicAdd(&gs[b], ge);
    for (int f = lane; f < 256; f += 32)
        atomicAdd(&pooled[(size_t)b * 256 + f], ge * h3[(size_t)n * 256 + f]);
}

// ---- head MLP: out = relu((pooled/gs) @ f1w + f1b) @ f2w + f2b, 64 graphs ----
__global__ void head_kernel(const float* __restrict__ pooled, const float* __restrict__ gs,
                            const float* __restrict__ f1w, const float* __restrict__ f1b,
                            const float* __restrict__ f2w, const float* __restrict__ f2b,
                            float* __restrict__ out) {
    int g = threadIdx.x;
    if (g >= N_GRAPHS) return;
    float inv = 1.0f / (gs[g] + 1e-16f);
    float o = f2b[0];
    for (int j = 0; j < 16; ++j) {
        float a = f1b[j];
        for (int f = 0; f < 256; ++f)
            a += pooled[(size_t)g * 256 + f] * inv * f1w[f * 16 + j];
        a = fmaxf(a, 0.0f);
        o += a * f2w[j];
    }
    out[g] = o;
}

// ------------------------------------------------------------------
static void run_gat_layer(const float* X, int K, int C,
                          const float* W, const float* as_, const float* ad_,
                          const float* bias, const float* gamma, const float* beta,
                          const float* mean, const float* var,
                          const int* ei,
                          float* Hbuf, float* accbuf, float* outbuf,
                          float* als, float* ald, unsigned* mkey, float* ssum,
                          hipStream_t stream) {
    int F = HEADS * C;
    int strips = cdiv_i(N_NODES, 16) * (F / 64);
    gemm_wmma_f32<<<cdiv_i(strips, 8), 256, 0, stream>>>(X, W, Hbuf, N_NODES, K, F);
    attn_logits<<<cdiv_i(N_NODES * HEADS, 256), 256, 0, stream>>>(Hbuf, as_, ad_, als, ald, C);
    gnn_zero<<<cdiv_i(N_NODES * F, 256), 256, 0, stream>>>(accbuf, N_NODES * F);
    gnn_zero<<<cdiv_i(N_NODES * HEADS, 256), 256, 0, stream>>>((float*)mkey, N_NODES * HEADS);
    gnn_zero<<<cdiv_i(N_NODES * HEADS, 256), 256, 0, stream>>>(ssum, N_NODES * HEADS);
    edge_max<<<cdiv_i(TOT_EDGES * HEADS, 256), 256, 0, stream>>>(ei, als, ald, mkey);
    edge_accum<<<cdiv_i(TOT_EDGES, 8), 256, 0, stream>>>(ei, als, ald, mkey, ssum, Hbuf, accbuf, C);
    gat_finalize<<<cdiv_i(N_NODES * F, 256), 256, 0, stream>>>(accbuf, ssum, bias, gamma, beta,
                                                               mean, var, outbuf, C);
}

extern "C" void kernel_launch(void* const* d_in, const int* in_sizes, int n_in,
                              void* d_out, int out_size, void* d_ws, size_t ws_size,
                              hipStream_t stream) {
    const float* x     = (const float*)d_in[0];
    const int*   ei    = (const int*)d_in[1];
    const int*   batch = (const int*)d_in[2];
    const float* W1  = (const float*)d_in[3];
    const float* a1s = (const float*)d_in[4];
    const float* a1d = (const float*)d_in[5];
    const float* b1  = (const float*)d_in[6];
    const float* g1  = (const float*)d_in[7];
    const float* be1 = (const float*)d_in[8];
    const float* m1  = (const float*)d_in[9];
    const float* v1  = (const float*)d_in[10];
    const float* W2  = (const float*)d_in[11];
    const float* a2s = (const float*)d_in[12];
    const float* a2d = (const float*)d_in[13];
    const float* b2  = (const float*)d_in[14];
    const float* g2  = (const float*)d_in[15];
    const float* be2 = (const float*)d_in[16];
    const float* m2  = (const float*)d_in[17];
    const float* v2  = (const float*)d_in[18];
    const float* W3  = (const float*)d_in[19];
    const float* a3s = (const float*)d_in[20];
    const float* a3d = (const float*)d_in[21];
    const float* b3  = (const float*)d_in[22];
    const float* g3  = (const float*)d_in[23];
    const float* be3 = (const float*)d_in[24];
    const float* m3  = (const float*)d_in[25];
    const float* v3  = (const float*)d_in[26];
    const float* gw  = (const float*)d_in[27];
    const float* gb  = (const float*)d_in[28];
    const float* f1w = (const float*)d_in[29];
    const float* f1b = (const float*)d_in[30];
    const float* f2w = (const float*)d_in[31];
    const float* f2b = (const float*)d_in[32];

    // ---- workspace carve-up (floats) ----
    char* p = (char*)d_ws;
    const size_t big = (size_t)N_NODES * 512 * sizeof(float);
    float* B0 = (float*)p; p += big;                 // layer input / finalized output
    float* B1 = (float*)p; p += big;                 // GEMM output (h)
    float* B2 = (float*)p; p += big;                 // edge accumulator
    float*    als  = (float*)p;    p += (size_t)N_NODES * HEADS * sizeof(float);
    float*    ald  = (float*)p;    p += (size_t)N_NODES * HEADS * sizeof(float);
    unsigned* mkey = (unsigned*)p; p += (size_t)N_NODES * HEADS * sizeof(unsigned);
    float*    ssum = (float*)p;    p += (size_t)N_NODES * HEADS * sizeof(float);
    float*    gate = (float*)p;    p += (size_t)N_NODES * sizeof(float);
    unsigned* gmk  = (unsigned*)p; p += (size_t)N_GRAPHS * sizeof(unsigned);
    float*    gs   = (float*)p;    p += (size_t)N_GRAPHS * sizeof(float);
    float*    pooled = (float*)p;  p += (size_t)N_GRAPHS * 256 * sizeof(float);
    (void)ws_size; (void)n_in; (void)in_sizes; (void)out_size;

    // Layer 1: 128 -> 8x32 = 256
    run_gat_layer(x, 128, 32, W1, a1s, a1d, b1, g1, be1, m1, v1,
                  ei, B1, B2, B0, als, ald, mkey, ssum, stream);
    // Layer 2: 256 -> 8x64 = 512
    run_gat_layer(B0, 256, 64, W2, a2s, a2d, b2, g2, be2, m2, v2,
                  ei, B1, B2, B0, als, ald, mkey, ssum, stream);
    // Layer 3: 512 -> 8x32 = 256
    run_gat_layer(B0, 512, 32, W3, a3s, a3d, b3, g3, be3, m3, v3,
                  ei, B1, B2, B0, als, ald, mkey, ssum, stream);

    // Global attention pooling
    gnn_zero<<<1, 256, 0, stream>>>((float*)gmk, N_GRAPHS);
    gnn_zero<<<1, 256, 0, stream>>>(gs, N_GRAPHS);
    gnn_zero<<<cdiv_i(N_GRAPHS * 256, 256), 256, 0, stream>>>(pooled, N_GRAPHS * 256);
    gate_kernel<<<cdiv_i(N_NODES, 8), 256, 0, stream>>>(B0, gw, gb, batch, gate, gmk);
    pool_kernel<<<cdiv_i(N_NODES, 8), 256, 0, stream>>>(B0, gate, gmk, batch, gs, pooled);
    head_kernel<<<1, 64, 0, stream>>>(pooled, gs, f1w, f1b, f2w, f2b, (float*)d_out);
}